// HeteroGCN_7928509628986
// MI455X (gfx1250) — compile-verified
//
#include <hip/hip_runtime.h>
#include <hip/hip_bf16.h>

typedef float v2f __attribute__((ext_vector_type(2)));
typedef float v8f __attribute__((ext_vector_type(8)));

#define D 64
#define N_DRUG 10000
#define N_GENE 50000

// ---------------------------------------------------------------------------
// K=64 tile MAC on the f32 WMMA path: C += (scale * A)(16x64) * B(64x16 slice)
// A row-major stride 64, B row-major stride 64. Per the ISA 16x16x4 f32
// layout each lane's A fragment is one matrix row (m = lane&15), so a per-row
// scale (the 1/degree of SAGE mean aggregation) is a single per-lane scalar
// fused into the A load — this deletes the separate mean-division pass.
// ---------------------------------------------------------------------------
__device__ __forceinline__ v8f tile_mac64(const float* __restrict__ A,
                                          const float* __restrict__ B,
                                          float scale, v8f c, int mn,
                                          int half) {
#pragma unroll
  for (int k = 0; k < D; k += 4) {
    const int kk = k + 2 * half;
    v2f a, b;
    a.x = A[mn * D + kk] * scale;
    a.y = A[mn * D + kk + 1] * scale;
    b.x = B[kk * D + mn];
    b.y = B[(kk + 1) * D + mn];
    c = __builtin_amdgcn_wmma_f32_16x16x4_f32(false, a, false, b, (short)0, c,
                                              false, false);
  }
  return c;
}

// ---------------------------------------------------------------------------
// h_drug = x_drug @ W_proj + b_proj   (M x 256) @ (256 x 64)
// One 16x16 output tile per wave32.
// ---------------------------------------------------------------------------
__global__ __launch_bounds__(256) void proj_kernel(
    const float* __restrict__ X, const float* __restrict__ W,
    const float* __restrict__ b, float* __restrict__ out, int M) {
  const int wave = (blockIdx.x * blockDim.x + threadIdx.x) >> 5;
  const int total = (M / 16) * (D / 16);
  if (wave >= total) return;  // wave-uniform: EXEC stays all-1s
  const int tm = wave >> 2, tn = wave & 3;
  const int lane = threadIdx.x & 31;
  const int half = lane >> 4, mn = lane & 15;

  const float* A = X + (size_t)(tm * 16) * 256;
  const float* B = W + tn * 16;
  v8f c = {};
#pragma unroll 4
  for (int k = 0; k < 256; k += 4) {
    const int kk = k + 2 * half;
    v2f a, bb;
    a.x = A[mn * 256 + kk];
    a.y = A[mn * 256 + kk + 1];
    bb.x = B[kk * D + mn];
    bb.y = B[(kk + 1) * D + mn];
    c = __builtin_amdgcn_wmma_f32_16x16x4_f32(false, a, false, bb, (short)0, c,
                                              false, false);
  }
  const float bias = b[tn * 16 + mn];
  float* O = out + (size_t)(tm * 16) * D + tn * 16;
#pragma unroll
  for (int v = 0; v < 8; ++v) O[(v + 8 * half) * D + mn] = c[v] + bias;
}

// ---------------------------------------------------------------------------
// One wave32 per edge: gather 64-float source row (float2/lane, one coalesced
// 256B transaction), scatter-add via global_atomic_add_f32 (L2-resident).
// ---------------------------------------------------------------------------
__global__ __launch_bounds__(256) void edge_scatter(
    const float* __restrict__ hsrc, const int* __restrict__ src,
    const int* __restrict__ dst, float* __restrict__ acc, int E) {
  const int e = (blockIdx.x * blockDim.x + threadIdx.x) >> 5;
  if (e >= E) return;
  const int lane = threadIdx.x & 31;
  const int s = src[e], d = dst[e];
  const v2f v = *(const v2f*)(hsrc + (size_t)s * D + lane * 2);
  float* o = acc + (size_t)d * D + lane * 2;
  atomicAdd(o, v.x);
  atomicAdd(o + 1, v.y);
}

// degree histogram (edge lists are layer-invariant: computed once per call)
__global__ __launch_bounds__(256) void count_kernel(const int* __restrict__ dst,
                                                    float* __restrict__ cnt,
                                                    int E) {
  const int i = blockIdx.x * blockDim.x + threadIdx.x;
  if (i < E) atomicAdd(cnt + dst[i], 1.0f);
}

// cnt[i] = 1 / max(cnt[i], 1)  (stored reciprocal, consumed by combine)
__global__ __launch_bounds__(256) void inv_kernel(float* __restrict__ cnt,
                                                  int n) {
  const int i = blockIdx.x * blockDim.x + threadIdx.x;
  if (i < n) cnt[i] = __frcp_rn(fmaxf(cnt[i], 1.0f));
}

// ---------------------------------------------------------------------------
// out = (inv1.A1)@B1 + A2@B2 [+ (inv3.A3)@B3 + A4@B4] + bias1 [+ bias2],
// optional ReLU. Up to 4 chained K=64 WMMA GEMMs per 16x16 tile (SAGE:
// mean@Wl + h@Wr per edge type, summed). Null checks are scalar-uniform.
// ---------------------------------------------------------------------------
__global__ __launch_bounds__(256) void combine_kernel(
    float* __restrict__ out, int M, const float* __restrict__ A1,
    const float* __restrict__ B1, const float* __restrict__ inv1,
    const float* __restrict__ A2, const float* __restrict__ B2,
    const float* __restrict__ A3, const float* __restrict__ B3,
    const float* __restrict__ inv3, const float* __restrict__ A4,
    const float* __restrict__ B4, const float* __restrict__ bias1,
    const float* __restrict__ bias2, int relu) {
  const int wave = (blockIdx.x * blockDim.x + threadIdx.x) >> 5;
  const int total = (M / 16) * (D / 16);
  if (wave >= total) return;
  const int tm = wave >> 2, tn = wave & 3;
  const int lane = threadIdx.x & 31;
  const int half = lane >> 4, mn = lane & 15;

  const size_t arow = (size_t)(tm * 16) * D;
  const float s1 = inv1[tm * 16 + mn];  // per-row 1/deg, one lane scalar
  v8f c = {};
  c = tile_mac64(A1 + arow, B1 + tn * 16, s1, c, mn, half);
  c = tile_mac64(A2 + arow, B2 + tn * 16, 1.0f, c, mn, half);
  if (A3) {
    const float s3 = inv3[tm * 16 + mn];
    c = tile_mac64(A3 + arow, B3 + tn * 16, s3, c, mn, half);
    c = tile_mac64(A4 + arow, B4 + tn * 16, 1.0f, c, mn, half);
  }

  float bias = bias1[tn * 16 + mn];
  if (bias2) bias += bias2[tn * 16 + mn];
  float* O = out + arow + tn * 16;
#pragma unroll
  for (int v = 0; v < 8; ++v) {
    float val = c[v] + bias;
    if (relu) val = fmaxf(val, 0.0f);
    O[(v + 8 * half) * D + mn] = val;
  }
}

__global__ __launch_bounds__(256) void zero_kernel(float* __restrict__ p,
                                                   long n4) {
  const long i = (long)blockIdx.x * blockDim.x + threadIdx.x;
  if (i < n4) *(float4*)(p + i * 4) = make_float4(0.f, 0.f, 0.f, 0.f);
}

// ---------------------------------------------------------------------------
extern "C" void kernel_launch(void* const* d_in, const int* in_sizes, int n_in,
                              void* d_out, int out_size, void* d_ws,
                              size_t ws_size, hipStream_t stream) {
  const float* x_drug = (const float*)d_in[0];
  const float* emb_gene = (const float*)d_in[1];
  const float* W_proj = (const float*)d_in[2];
  const float* b_proj = (const float*)d_in[3];
  // layer weights: [Wl, bl, Wr] x (dg, gd, gg) x 2 layers, starting at 4
  const float* Wl[2][3];
  const float* bl[2][3];
  const float* Wr[2][3];
  int idx = 4;
  for (int l = 0; l < 2; ++l)
    for (int t = 0; t < 3; ++t) {  // t: 0=dg 1=gd 2=gg
      Wl[l][t] = (const float*)d_in[idx++];
      bl[l][t] = (const float*)d_in[idx++];
      Wr[l][t] = (const float*)d_in[idx++];
    }
  const int* dg_src = (const int*)d_in[22];
  const int* dg_dst = (const int*)d_in[23];
  const int* gd_src = (const int*)d_in[24];
  const int* gd_dst = (const int*)d_in[25];
  const int* gg_src = (const int*)d_in[26];
  const int* gg_dst = (const int*)d_in[27];
  const int E_DG = in_sizes[22], E_GD = in_sizes[24], E_GG = in_sizes[26];

  // workspace layout (floats)
  float* ws = (float*)d_ws;
  float* h_drugA = ws;                       // 640000  (proj output)
  float* h_drugB = h_drugA + N_DRUG * D;     // 640000  (layer-0 drug out)
  float* h_geneA = h_drugB + N_DRUG * D;     // 3200000 (layer-0 gene out)
  float* acc_dg = h_geneA + N_GENE * D;      // 3200000
  float* acc_gg = acc_dg + N_GENE * D;       // 3200000
  float* acc_gd = acc_gg + N_GENE * D;       // 640000
  float* cnt_dg = acc_gd + N_DRUG * D;       // 50000 (holds 1/deg after inv)
  float* cnt_gg = cnt_dg + N_GENE;           // 50000
  float* cnt_gd = cnt_gg + N_GENE;           // 10000
  const long acc_floats = (long)2 * N_GENE * D + (long)N_DRUG * D;  // 7.04M
  const long cnt_floats = 2 * N_GENE + N_DRUG;                      // 110k

  float* out_drug = (float*)d_out;
  float* out_gene = (float*)d_out + (size_t)N_DRUG * D;

  const dim3 blk(256);
  const int WPB = 256 / 32;  // waves per block

  // ---- one-time: degrees -> reciprocal (edge lists identical across layers)
  {
    long n4 = (cnt_floats + 3) / 4;
    zero_kernel<<<dim3((unsigned)((n4 + 255) / 256)), blk, 0, stream>>>(cnt_dg,
                                                                        n4);
    count_kernel<<<dim3((E_DG + 255) / 256), blk, 0, stream>>>(dg_dst, cnt_dg,
                                                               E_DG);
    count_kernel<<<dim3((E_GG + 255) / 256), blk, 0, stream>>>(gg_dst, cnt_gg,
                                                               E_GG);
    count_kernel<<<dim3((E_GD + 255) / 256), blk, 0, stream>>>(gd_dst, cnt_gd,
                                                               E_GD);
    inv_kernel<<<dim3((unsigned)((cnt_floats + 255) / 256)), blk, 0, stream>>>(
        cnt_dg, (int)cnt_floats);
  }

  // ---- input projection: h_drugA = x_drug @ W_proj + b_proj
  {
    int waves = (N_DRUG / 16) * (D / 16);
    proj_kernel<<<dim3((waves + WPB - 1) / WPB), blk, 0, stream>>>(
        x_drug, W_proj, b_proj, h_drugA, N_DRUG);
  }

  const float* h_drug = h_drugA;
  const float* h_gene = emb_gene;

  for (int l = 0; l < 2; ++l) {
    const int last = (l == 1);
    // zero accumulators (contiguous region)
    {
      long n4 = (acc_floats + 3) / 4;
      zero_kernel<<<dim3((unsigned)((n4 + 255) / 256)), blk, 0, stream>>>(
          acc_dg, n4);
    }
    // scatter phase
    edge_scatter<<<dim3((unsigned)((E_DG + WPB - 1) / WPB)), blk, 0, stream>>>(
        h_drug, dg_src, dg_dst, acc_dg, E_DG);
    edge_scatter<<<dim3((unsigned)((E_GG + WPB - 1) / WPB)), blk, 0, stream>>>(
        h_gene, gg_src, gg_dst, acc_gg, E_GG);
    edge_scatter<<<dim3((unsigned)((E_GD + WPB - 1) / WPB)), blk, 0, stream>>>(
        h_gene, gd_src, gd_dst, acc_gd, E_GD);
    // combine (WMMA, mean-scale fused into A loads):
    // gene = mean_dg@Wl_dg + h_gene@Wr_dg + mean_gg@Wl_gg + h_gene@Wr_gg + b
    float* ng = last ? out_gene : h_geneA;
    float* nd = last ? out_drug : h_drugB;
    {
      int waves = (N_GENE / 16) * (D / 16);
      combine_kernel<<<dim3((waves + WPB - 1) / WPB), blk, 0, stream>>>(
          ng, N_GENE, acc_dg, Wl[l][0], cnt_dg, h_gene, Wr[l][0], acc_gg,
          Wl[l][2], cnt_gg, h_gene, Wr[l][2], bl[l][0], bl[l][2],
          last ? 0 : 1);
    }
    // drug = mean_gd@Wl_gd + h_drug@Wr_gd + bl_gd
    {
      int waves = (N_DRUG / 16) * (D / 16);
      combine_kernel<<<dim3((waves + WPB - 1) / WPB), blk, 0, stream>>>(
          nd, N_DRUG, acc_gd, Wl[l][1], cnt_gd, h_drug, Wr[l][1], nullptr,
          nullptr, nullptr, nullptr, nullptr, bl[l][1], nullptr, last ? 0 : 1);
    }
    h_drug = nd;
    h_gene = ng;
  }
}